// RC_Module_43473658970737
// MI455X (gfx1250) — compile-verified
//
#include <hip/hip_runtime.h>
#include <hip/hip_bf16.h>
#include <math.h>

// v2f = 16x4 F32 A-fragment / 4x16 F32 B-fragment (2 VGPRs/lane, wave32)
typedef __attribute__((ext_vector_type(2))) float v2f;
// v8f = 16x16 F32 C/D fragment (8 VGPRs/lane, wave32)
typedef __attribute__((ext_vector_type(8))) float v8f;

#define MLP_P 49          // 7*7 positions (= K of collapsed GEMM)
#define NCHUNK 13         // ceil(49/4) K-chunks of the 16x16x4 WMMA
#define COUT 64
#define DOUT 4

// ---------------------------------------------------------------------------
// Prep kernel (1 block): fold the two-layer MLP into a single 49x4 matrix
//   A'[p][d] = (1/49) * sum_o W1[p,0,o] * W2[p,o,d]
//   C [p][d] =          sum_o b1[p,o]   * W2[p,o,d] + b2[p,d]
// and emit:
//   wsB : 13 chunks x 2 VGPRs x 32 lanes, the B-fragment (4x16 F32) per chunk
//         B[k][n] = A'[k][n] for n<4, k<49 else 0
//         layout: wsB[chunk*64 + r*32 + lane], k = chunk*4 + (lane>>4)*2 + r,
//                 n = lane & 15
//   wsC : 16 floats, wsC[n] = mean_p C[p][n] for n<4 else 0  (folded into
//         the WMMA C accumulator, broadcast down column n)
// ---------------------------------------------------------------------------
__global__ void __launch_bounds__(256)
mlp49_prep(const float* __restrict__ W1, const float* __restrict__ b1,
           const float* __restrict__ W2, const float* __restrict__ b2,
           float* __restrict__ wsB, float* __restrict__ wsC) {
    __shared__ float sA[MLP_P][DOUT];
    __shared__ float sC[MLP_P][DOUT];
    const int t = threadIdx.x;

    if (t < MLP_P * DOUT) {
        const int p = t >> 2;
        const int d = t & 3;
        float a = 0.0f, c = 0.0f;
        #pragma unroll 8
        for (int o = 0; o < COUT; ++o) {
            const float w2 = W2[p * (COUT * DOUT) + o * DOUT + d];
            a = fmaf(W1[p * COUT + o], w2, a);
            c = fmaf(b1[p * COUT + o], w2, c);
        }
        sA[p][d] = a * (1.0f / 49.0f);
        sC[p][d] = c + b2[p * DOUT + d];
    }
    __syncthreads();

    // B fragments: 13*64 entries, directly in per-lane WMMA layout
    for (int i = t; i < NCHUNK * 64; i += blockDim.x) {
        const int chunk = i >> 6;
        const int r     = (i >> 5) & 1;
        const int lane  = i & 31;
        const int n     = lane & 15;
        const int k     = chunk * 4 + ((lane >> 4) << 1) + r;
        float v = 0.0f;
        if (n < DOUT && k < MLP_P) v = sA[k][n];
        wsB[i] = v;
    }
    // column bias, broadcast table indexed by (lane & 15)
    if (t < 16) {
        float c = 0.0f;
        if (t < DOUT) {
            for (int p = 0; p < MLP_P; ++p) c += sC[p][t];
            c *= (1.0f / 49.0f);
        }
        wsC[t] = c;
    }
}

// ---------------------------------------------------------------------------
// Main streaming kernel: one wave32 handles 16 batch rows.
//   acc(16x16) = X_tile(16x52, zero-padded) * B(52x16)  via 13x
//   V_WMMA_F32_16X16X4_F32, C preloaded with the column bias. Columns 0..3 of
//   D are the live outputs; epilogue = tanh -> RNE round quantizer.
// Memory-bound: 12.8 MB x-read + 1 MB out-write ~= 0.6 us at 23.3 TB/s.
// ---------------------------------------------------------------------------
__global__ void __launch_bounds__(256)
mlp49_wmma(const float* __restrict__ x, const float* __restrict__ wsB,
           const float* __restrict__ wsC, float* __restrict__ out, int B) {
    const int lane = threadIdx.x & 31;
    const int wid  = (blockIdx.x * blockDim.x + threadIdx.x) >> 5;
    const int row_base = wid << 4;
    if (row_base >= B) return;            // wave-uniform; EXEC stays all-1s

    const int n  = lane & 15;             // column of B / D this lane owns
    const int hi = lane >> 4;             // half-wave select (K / M offset)

    // B fragments: identical for every wave -> stays L2-resident (192 MB L2)
    v2f bfrag[NCHUNK];
    #pragma unroll
    for (int c = 0; c < NCHUNK; ++c) {
        bfrag[c].x = wsB[c * 64 + lane];
        bfrag[c].y = wsB[c * 64 + 32 + lane];
    }

    // fold bias into the accumulator: C[m][n] = bias[n] for all m
    const float cbias = wsC[n];
    v8f acc;
    #pragma unroll
    for (int v = 0; v < 8; ++v) acc[v] = cbias;

    // A fragment source row for this lane (clamped for ragged tail)
    int arow = row_base + n;
    if (arow >= B) arow = B - 1;          // duplicate row; stores are guarded
    const float* __restrict__ xrow = x + (size_t)arow * MLP_P;

    // chunks 0..11: K in [0,48) -- always in range
    #pragma unroll
    for (int c = 0; c < NCHUNK - 1; ++c) {
        const int k0 = c * 4 + (hi << 1);
        v2f a;
        a.x = xrow[k0];
        a.y = xrow[k0 + 1];
        acc = __builtin_amdgcn_wmma_f32_16x16x4_f32(
            false, a, false, bfrag[c], (short)0, acc, false, false);
    }
    // chunk 12: K = 48..51, only K=48 (lanes 0-15, r=0) is real data
    {
        v2f a;
        a.x = (hi == 0) ? xrow[48] : 0.0f;
        a.y = 0.0f;
        acc = __builtin_amdgcn_wmma_f32_16x16x4_f32(
            false, a, false, bfrag[NCHUNK - 1], (short)0, acc, false, false);
    }

    // Epilogue + store: lanes with n<4 own D columns 0..3 (the real outputs)
    if (n < DOUT) {
        #pragma unroll
        for (int v = 0; v < 8; ++v) {
            const int row = row_base + v + (hi << 3);
            if (row < B) {
                float y = tanhf(acc[v]);
                y = rintf(y * 127.0f);            // ste_round (RNE = jnp.round)
                y = y + 127.0f;
                y = fminf(fmaxf(y, 0.0f), 255.0f);
                y = rintf(y);                     // second ste_round (identity on ints)
                out[(size_t)row * DOUT + n] = y * (1.0f / 255.0f);
            }
        }
    }
}

extern "C" void kernel_launch(void* const* d_in, const int* in_sizes, int n_in,
                              void* d_out, int out_size, void* d_ws, size_t ws_size,
                              hipStream_t stream) {
    const float* x  = (const float*)d_in[0];   // [B,7,7,1]
    const float* W1 = (const float*)d_in[1];   // [49,1,64]
    const float* b1 = (const float*)d_in[2];   // [49,64]
    const float* W2 = (const float*)d_in[3];   // [49,64,4]
    const float* b2 = (const float*)d_in[4];   // [49,4]
    float* out = (float*)d_out;                // [B,4,1,1]

    float* wsB = (float*)d_ws;                 // 13*64 = 832 floats
    float* wsC = wsB + NCHUNK * 64;            // 16 floats

    const int B = in_sizes[0] / MLP_P;         // x elements = B*49 (Cin=1)

    mlp49_prep<<<1, 256, 0, stream>>>(W1, b1, W2, b2, wsB, wsC);

    const int waves   = (B + 15) >> 4;         // 16 rows per wave32
    const int threads = 256;                   // 8 waves/block (WGP-friendly)
    const int blocks  = (waves * 32 + threads - 1) / threads;
    mlp49_wmma<<<blocks, threads, 0, stream>>>(x, wsB, wsC, out, B);
}